// MyModel_6124623364543
// MI455X (gfx1250) — compile-verified
//
#include <hip/hip_runtime.h>

#define NN   50000
#define NE   800000
#define INF_ 128
#define HID  64
#define HID4 16     // HID/4 quads per row
#define DHID 384    // (D+1)*2*HID = 3*64*2

typedef float v2f __attribute__((ext_vector_type(2)));
typedef float v8f __attribute__((ext_vector_type(8)));

// ---------------- elementwise / graph kernels ----------------

__global__ void k_zero(float* __restrict__ p, int n) {
  int i = blockIdx.x * blockDim.x + threadIdx.x;
  if (i < n) p[i] = 0.0f;
}

__global__ void k_degree(const int* __restrict__ dst, float* __restrict__ deg, int e) {
  int i = blockIdx.x * blockDim.x + threadIdx.x;
  if (i < e) atomicAdd(&deg[dst[i]], 1.0f);
}

__global__ void k_dinv(float* __restrict__ d, int n) {
  int i = blockIdx.x * blockDim.x + threadIdx.x;
  if (i < n) {
    float v = d[i];
    v = v < 1.0f ? 1.0f : v;
    d[i] = rsqrtf(v);
  }
}

// t = f * dinv[row]; agg = 0   (float4 per thread)
__global__ void k_scale_zero4(const float4* __restrict__ f, const float* __restrict__ dinv,
                              float4* __restrict__ t, float4* __restrict__ agg) {
  int i = blockIdx.x * blockDim.x + threadIdx.x;
  if (i >= NN * HID4) return;
  float dv = dinv[i >> 4];
  float4 v = f[i];
  v.x *= dv; v.y *= dv; v.z *= dv; v.w *= dv;
  t[i] = v;
  agg[i] = make_float4(0.f, 0.f, 0.f, 0.f);
}

// agg[dst][q*4..q*4+3] += t[src][q*4..q*4+3]  (one thread per edge-quad)
__global__ void k_scatter4(const int* __restrict__ src, const int* __restrict__ dst,
                           const float4* __restrict__ t, float* __restrict__ agg) {
  int i = blockIdx.x * blockDim.x + threadIdx.x;
  if (i >= NE * HID4) return;
  int e = i >> 4;
  int q = i & 15;
  float4 v = t[(size_t)src[e] * HID4 + q];
  float* a = agg + (size_t)dst[e] * HID + q * 4;
  atomicAdd(a + 0, v.x);
  atomicAdd(a + 1, v.y);
  atomicAdd(a + 2, v.z);
  atomicAdd(a + 3, v.w);
}

// f = f - agg * dinv[row]; out[row, cb + j*64 + col] += coef_j * f  (float4, fused)
__global__ void k_update_combine4(float4* __restrict__ f, const float4* __restrict__ agg,
                                  const float* __restrict__ dinv, float* __restrict__ out,
                                  int col_base, float c0, float c1, float c2) {
  int i = blockIdx.x * blockDim.x + threadIdx.x;
  if (i >= NN * HID4) return;
  int row = i >> 4;
  int q   = i & 15;
  float dv = dinv[row];
  float4 fv = f[i];
  float4 av = agg[i];
  fv.x -= av.x * dv; fv.y -= av.y * dv; fv.z -= av.z * dv; fv.w -= av.w * dv;
  f[i] = fv;
  float4* o = (float4*)(out + (size_t)row * DHID + col_base + q * 4);
  const float cs[3] = {c0, c1, c2};
#pragma unroll
  for (int j = 0; j < 3; ++j) {
    float4 ov = o[j * HID4];
    ov.x += cs[j] * fv.x; ov.y += cs[j] * fv.y;
    ov.z += cs[j] * fv.z; ov.w += cs[j] * fv.w;
    o[j * HID4] = ov;
  }
}

// ---------------- WMMA GEMM: f0 = relu(X[NNx128] @ W[128x64] + b), fused stage-0 combine ----
// 1 block = 16 rows; 4 waves, one 16x16 N-tile each. K-loop of v_wmma_f32_16x16x4_f32.

__global__ __launch_bounds__(128) void k_gemm1_relu_combine(
    const float* __restrict__ X, const float* __restrict__ W,
    const float* __restrict__ bias, float* __restrict__ Y,
    float* __restrict__ out, int col_base, float c0, float c1, float c2) {
  const int lane = threadIdx.x & 31;
  const int wave = threadIdx.x >> 5;
  const int row0 = blockIdx.x * 16;
  const int n0   = wave * 16;
  const int mr   = lane & 15;          // M for A-frag, N for B-frag / D-frag
  const int kq   = (lane >> 4) << 1;   // K pair base: 0 or 2
  v8f c = {};
  const float* xrow = X + (size_t)(row0 + mr) * INF_;
  for (int k = 0; k < INF_; k += 4) {
    v2f a, b;
    a.x = xrow[k + kq];
    a.y = xrow[k + kq + 1];
    b.x = W[(size_t)(k + kq) * HID + n0 + mr];
    b.y = W[(size_t)(k + kq + 1) * HID + n0 + mr];
    c = __builtin_amdgcn_wmma_f32_16x16x4_f32(false, a, false, b, (short)0, c, false, false);
  }
  const int mhi = (lane >> 4) << 3;
  const float bv = bias[n0 + mr];
#pragma unroll
  for (int r = 0; r < 8; ++r) {
    float v = c[r] + bv;
    v = v > 0.0f ? v : 0.0f;
    const size_t row = (size_t)(row0 + mhi + r);
    Y[row * HID + n0 + mr] = v;
    float* o = out + row * DHID + col_base + n0 + mr;
    o[0]       = c0 * v;
    o[HID]     = c1 * v;
    o[2 * HID] = c2 * v;
  }
}

// ---------------- Head: logits = relu(H[NNx384] @ Wm1[384x64] + bm1) @ Wm2[64x2] + bm2 ----

__global__ __launch_bounds__(128) void k_head(
    const float* __restrict__ H,
    const float* __restrict__ Wm1, const float* __restrict__ bm1,
    const float* __restrict__ Wm2, const float* __restrict__ bm2,
    float* __restrict__ logits) {
  __shared__ float m[16][HID];
  const int lane = threadIdx.x & 31;
  const int wave = threadIdx.x >> 5;
  const int row0 = blockIdx.x * 16;
  const int n0   = wave * 16;
  const int mr   = lane & 15;
  const int kq   = (lane >> 4) << 1;
  v8f c = {};
  const float* hrow = H + (size_t)(row0 + mr) * DHID;
  for (int k = 0; k < DHID; k += 4) {
    v2f a, b;
    a.x = hrow[k + kq];
    a.y = hrow[k + kq + 1];
    b.x = Wm1[(size_t)(k + kq) * HID + n0 + mr];
    b.y = Wm1[(size_t)(k + kq + 1) * HID + n0 + mr];
    c = __builtin_amdgcn_wmma_f32_16x16x4_f32(false, a, false, b, (short)0, c, false, false);
  }
  const int mhi = (lane >> 4) << 3;
  const float bv = bm1[n0 + mr];
#pragma unroll
  for (int r = 0; r < 8; ++r) {
    float v = c[r] + bv;
    m[mhi + r][n0 + mr] = v > 0.0f ? v : 0.0f;
  }
  __syncthreads();
  if (threadIdx.x < 32) {
    int row = threadIdx.x >> 1;
    int col = threadIdx.x & 1;
    float acc = bm2[col];
#pragma unroll
    for (int k = 0; k < HID; ++k) acc += m[row][k] * Wm2[k * 2 + col];
    logits[(size_t)(row0 + row) * 2 + col] = acc;
  }
}

// ---------------- host-side branch driver ----------------

static void run_branch(const float* X, const int* src, const int* dst,
                       const float* W1, const float* b1,
                       float* out, int col_base,
                       const float sc[3][3],  // sc[k][j] = coef of f_k for filter j
                       float* f, float* t, float* agg, float* dinv,
                       hipStream_t stream) {
  const int nNodeB = (NN + 255) / 256;
  const int nEdgeB = (NE + 255) / 256;
  const int nQuadB = (NN * HID4 + 255) / 256;   // node-quads
  const int nScatB = (NE * HID4 + 255) / 256;   // edge-quads

  k_zero  <<<nNodeB, 256, 0, stream>>>(dinv, NN);
  k_degree<<<nEdgeB, 256, 0, stream>>>(dst, dinv, NE);
  k_dinv  <<<nNodeB, 256, 0, stream>>>(dinv, NN);

  k_gemm1_relu_combine<<<NN / 16, 128, 0, stream>>>(X, W1, b1, f, out, col_base,
                                                    sc[0][0], sc[0][1], sc[0][2]);

  for (int step = 1; step <= 2; ++step) {
    k_scale_zero4<<<nQuadB, 256, 0, stream>>>((const float4*)f, dinv,
                                              (float4*)t, (float4*)agg);
    k_scatter4   <<<nScatB, 256, 0, stream>>>(src, dst, (const float4*)t, agg);
    k_update_combine4<<<nQuadB, 256, 0, stream>>>((float4*)f, (const float4*)agg, dinv,
                                                  out, col_base,
                                                  sc[step][0], sc[step][1], sc[step][2]);
  }
}

extern "C" void kernel_launch(void* const* d_in, const int* in_sizes, int n_in,
                              void* d_out, int out_size, void* d_ws, size_t ws_size,
                              hipStream_t stream) {
  (void)in_sizes; (void)n_in; (void)out_size; (void)ws_size;
  const float* x     = (const float*)d_in[0];
  const float* sim_x = (const float*)d_in[1];
  const int*   src   = (const int*)d_in[2];
  const int*   dst   = (const int*)d_in[3];
  const int*   ssrc  = (const int*)d_in[4];
  const int*   sdst  = (const int*)d_in[5];
  const float* W1o   = (const float*)d_in[6];
  const float* b1o   = (const float*)d_in[7];
  const float* W1s   = (const float*)d_in[8];
  const float* b1s   = (const float*)d_in[9];
  const float* Wm1   = (const float*)d_in[10];
  const float* bm1   = (const float*)d_in[11];
  const float* Wm2   = (const float*)d_in[12];
  const float* bm2   = (const float*)d_in[13];

  float* out    = (float*)d_out;               // h_all: NN x 384
  float* logits = out + (size_t)NN * DHID;     // NN x 2

  float* f    = (float*)d_ws;                  // NN*64
  float* t    = f    + (size_t)NN * HID;       // NN*64
  float* agg  = t    + (size_t)NN * HID;       // NN*64
  float* dinv = agg  + (size_t)NN * HID;       // NN

  // thetas for d=2 (ascending): th0={3,-3,0.75}, th1={0,3,-1.5}, th2={0,0,0.75}
  // sc[k][j] = th[j][k] = coefficient multiplying f_k in filter j's output
  const float sc_o[3][3] = {
      {3.0f,   0.0f,  0.0f},
      {-3.0f,  3.0f,  0.0f},
      {0.75f, -1.5f,  0.75f}};
  const float sc_s[3][3] = {
      {4.0f, 4.0f, 4.0f},
      {4.0f, 4.0f, 4.0f},
      {4.0f, 4.0f, 4.0f}};

  run_branch(x,     src,  dst,  W1o, b1o, out,   0, sc_o, f, t, agg, dinv, stream);
  run_branch(sim_x, ssrc, sdst, W1s, b1s, out, 192, sc_s, f, t, agg, dinv, stream);

  k_head<<<NN / 16, 128, 0, stream>>>(out, Wm1, bm1, Wm2, bm2, logits);
}